// L1TV1DLayer_23536420782355
// MI455X (gfx1250) — compile-verified
//
#include <hip/hip_runtime.h>
#include <math.h>

// Problem constants from the reference: y is (8, 3, 256, 512) f32.
#define TVW      512          // row length
#define TVH      256          // H
#define TVC      3            // channels
#define ROWS     64           // rows (lanes) per workgroup = 2 waves
#define STRIDE   516          // padded LDS row stride in floats: 16B aligned,
                              // bank = (4*lane + k) % 64 -> <=2-way conflict

__device__ __forceinline__ void async_load16(unsigned lds_addr, const float* g) {
    unsigned long long gaddr = (unsigned long long)(const void*)g;
    asm volatile("global_load_async_to_lds_b128 %0, %1, off"
                 :: "v"(lds_addr), "v"(gaddr) : "memory");
}
__device__ __forceinline__ void async_store16(unsigned lds_addr, float* g) {
    unsigned long long gaddr = (unsigned long long)(void*)g;
    asm volatile("global_store_async_from_lds_b128 %0, %1, off"
                 :: "v"(gaddr), "v"(lds_addr) : "memory");
}

__global__ __launch_bounds__(ROWS)
void tv1d_condat_kernel(const float* __restrict__ y,
                        const float* __restrict__ lmbd,
                        float* __restrict__ out,
                        int nrows)
{
    extern __shared__ float smem[];                       // ROWS*STRIDE floats
    const int tid       = threadIdx.x;
    const int blockRow0 = blockIdx.x * ROWS;
    const bool fullTile = (blockRow0 + ROWS) <= nrows;    // uniform per block

    // Raw LDS byte address of the dynamic shared segment (no static LDS here).
    const unsigned ldsBase = __builtin_amdgcn_groupstaticsize();

    // ---------------- Phase 1: coalesced async HBM -> LDS ----------------
    // chunk c = 16 bytes = 4 floats; 8192 chunks per 64-row tile.
    const int chunks = ROWS * TVW / 4;
    const float* gbase = y + (size_t)blockRow0 * TVW;
    if (fullTile) {
        for (int c = tid; c < chunks; c += ROWS) {
            int row = c >> 7;                 // (4c)/512
            int col = (c & 127) << 2;         // (4c)%512
            async_load16(ldsBase + (unsigned)((row * STRIDE + col) * 4),
                         gbase + ((size_t)c << 2));
        }
    } else {
        for (int c = tid; c < chunks; c += ROWS) {
            int row = c >> 7;
            int col = (c & 127) << 2;
            if (blockRow0 + row < nrows)
                async_load16(ldsBase + (unsigned)((row * STRIDE + col) * 4),
                             gbase + ((size_t)c << 2));
        }
    }
    asm volatile("s_wait_asynccnt 0" ::: "memory");   // my wave's copies done
    __syncthreads();                                  // all waves' copies visible

    // ---------------- Phase 2: Condat TV1D prox, in place in LDS ----------
    const int gRow = blockRow0 + tid;
    if (gRow < nrows) {
        float* rowp = smem + tid * STRIDE;

        // lam = softplus(lmbd[channel]); channel = (row / H) % C
        int ch    = (gRow / TVH) % TVC;
        float lx  = lmbd[ch];
        float lam = (lx > 20.0f) ? lx : log1pf(expf(lx));

        int   k = 0, k0 = 0, km = 0, kp = 0;
        float umin = lam, umax = -lam;
        float vmin = rowp[0] - lam, vmax = rowp[0] + lam;

        for (;;) {
            if (k == TVW - 1) {                      // ---- branch A ----
                if (umin < 0.0f) {                   // a1: flush at vmin
                    for (int i = k0; i <= km; ++i) rowp[i] = vmin;
                    k = k0 = km = km + 1;            // kp unchanged
                    int rd = (k < TVW) ? k : TVW - 1;
                    float vn = rowp[rd];
                    umin = lam;
                    umax = vn + lam - vmax;          // vmax unchanged
                    vmin = vn;
                } else if (umax > 0.0f) {            // a2: flush at vmax
                    for (int i = k0; i <= kp; ++i) rowp[i] = vmax;
                    k = k0 = kp = kp + 1;            // km unchanged
                    int rd = (k < TVW) ? k : TVW - 1;
                    float vn = rowp[rd];
                    umin = vn - lam - vmin;          // vmin unchanged
                    umax = -lam;
                    vmax = vn;
                } else {                             // a3: terminate
                    float v = vmin + umin / (float)(k - k0 + 1);
                    for (int i = k0; i <= k; ++i) rowp[i] = v;
                    break;
                }
            } else {                                 // ---- branch B ----
                float yn     = rowp[k + 1];
                float umin_t = umin + yn - vmin;
                float umax_t = umax + yn - vmax;
                if (umin_t < -lam) {                 // b1: negative jump
                    for (int i = k0; i <= km; ++i) rowp[i] = vmin;
                    k = k0 = km = kp = km + 1;
                    int rd = (k < TVW) ? k : TVW - 1;
                    float vn = rowp[rd];
                    vmin = vn;
                    vmax = vn + 2.0f * lam;
                    umin = lam;  umax = -lam;
                } else if (umax_t > lam) {           // b2: positive jump
                    for (int i = k0; i <= kp; ++i) rowp[i] = vmax;
                    k = k0 = km = kp = kp + 1;
                    int rd = (k < TVW) ? k : TVW - 1;
                    float vn = rowp[rd];
                    vmax = vn;
                    vmin = vn - 2.0f * lam;
                    umin = lam;  umax = -lam;
                } else {                             // b3: extend segment
                    k += 1;
                    float denom = (float)(k - k0 + 1);
                    if (umin_t >= lam) { vmin += (umin_t - lam) / denom; umin = lam;  km = k; }
                    else               { umin = umin_t; }
                    if (umax_t <= -lam){ vmax += (umax_t + lam) / denom; umax = -lam; kp = k; }
                    else               { umax = umax_t; }
                }
            }
        }
    }
    __syncthreads();

    // ---------------- Phase 3: coalesced async LDS -> HBM -----------------
    float* obase = out + (size_t)blockRow0 * TVW;
    if (fullTile) {
        for (int c = tid; c < chunks; c += ROWS) {
            int row = c >> 7;
            int col = (c & 127) << 2;
            async_store16(ldsBase + (unsigned)((row * STRIDE + col) * 4),
                          obase + ((size_t)c << 2));
        }
    } else {
        for (int c = tid; c < chunks; c += ROWS) {
            int row = c >> 7;
            int col = (c & 127) << 2;
            if (blockRow0 + row < nrows)
                async_store16(ldsBase + (unsigned)((row * STRIDE + col) * 4),
                              obase + ((size_t)c << 2));
        }
    }
    asm volatile("s_wait_asynccnt 0" ::: "memory");
}

extern "C" void kernel_launch(void* const* d_in, const int* in_sizes, int n_in,
                              void* d_out, int out_size, void* d_ws, size_t ws_size,
                              hipStream_t stream) {
    const float* y    = (const float*)d_in[0];   // (8,3,256,512) f32
    const float* lmbd = (const float*)d_in[1];   // (1,3) f32
    float*       out  = (float*)d_out;

    const int nrows  = in_sizes[0] / TVW;        // 6144
    const int blocks = (nrows + ROWS - 1) / ROWS;
    const size_t shmem = (size_t)ROWS * STRIDE * sizeof(float); // ~129 KB of 320 KB/WGP

    tv1d_condat_kernel<<<blocks, ROWS, shmem, stream>>>(y, lmbd, out, nrows);
}